// BackProjNet_43198781063637
// MI455X (gfx1250) — compile-verified
//
#include <hip/hip_runtime.h>
#include <cstddef>

// ---- problem constants -------------------------------------------------
#define CHN     8
#define NVIEWS  128
#define NDET    368
#define NRAY    (NVIEWS * NDET)          // 47104
#define MTOT    (128 * 128 * NVIEWS)     // 2097152 gather positions
#define OC1     112                      // 14*CH
#define OC2     56                       // 7*CH
#define HPITCH  372                      // 368 + halo/pad, zero-initialized
#define XRECORD 64                       // padded AoS record: [c*8 + f], slot 7 zero
#define HFLOATS ((size_t)NVIEWS * OC1 * HPITCH)   // 5,332,992
#define XFLOATS ((size_t)NRAY * XRECORD)          // 3,014,656

typedef __attribute__((ext_vector_type(2))) float v2f;
typedef __attribute__((ext_vector_type(8))) float v8f;

// =======================================================================
// Kernel 1: conv1 (8 -> 112 ch, k=3, pad=1) + bias + exact GELU.
// One block per view.  Input staged into zero-padded LDS (halo branch-free),
// weights staged into LDS reordered k-major (r = k*8 + ic) so that every
// WMMA operand address is base + immediate offset.  One wave per 16x16
// (oc x position) tile, WMMA f32 16x16x4, K = 24 in 6 steps.
// Writes zero-padded h[view][oc][1+u] (pads stay 0 from memset).
// =======================================================================
__global__ __launch_bounds__(256) void k_conv1_gelu(
    const float* __restrict__ in,   // [CH][V][U]
    const float* __restrict__ w1,   // [112][8][3]
    const float* __restrict__ b1,   // [112]
    float* __restrict__ h)          // [V][112][372]
{
    __shared__ float sIn[CHN * HPITCH];   // [ic][1+u], pads zero
    __shared__ float sW[OC1 * 24];        // sW[oc*24 + k*8 + ic]
    __shared__ float sB[OC1];

    int tid  = threadIdx.x;
    int view = blockIdx.x;

    // stage input with zero halo (sIn[ic][p], p = 1+u)
    for (int i = tid; i < CHN * HPITCH; i += 256) {
        int ic = i / HPITCH, p = i - ic * HPITCH;
        int u = p - 1;
        sIn[i] = (u >= 0 && u < NDET)
                   ? in[ic * (NVIEWS * NDET) + view * NDET + u] : 0.f;
    }
    // stage weights reordered k-major
    for (int i = tid; i < OC1 * 24; i += 256) {
        int oc = i / 24, r = i - oc * 24;
        int k = r >> 3, ic = r & 7;
        sW[i] = w1[oc * 24 + ic * 3 + k];
    }
    for (int i = tid; i < OC1; i += 256) sB[i] = b1[i];
    __syncthreads();

    int lane   = tid & 31;
    int waveId = tid >> 5;
    int row = lane & 15;                  // M (A) / N (B) index
    int hi  = lane >> 4;                  // K-half select
    int rA  = hi << 1;

    for (int t = waveId; t < 7 * 23; t += 8) {
        int ocT = t / 23, pT = t - ocT * 23;
        int p0  = pT << 4;
        int oc  = (ocT << 4) + row;

        const float* aBase = &sW[oc * 24 + rA];
        const float* bBase = &sIn[rA * HPITCH + p0 + row];
        const float* cBase = &sB[(ocT << 4) + (hi << 3)];

        v8f acc;
#pragma unroll
        for (int j = 0; j < 8; ++j) acc[j] = cBase[j];

#pragma unroll
        for (int step = 0; step < 6; ++step) {
            int kb  = step << 2;
            int icc = kb & 7;             // 0 or 4 (compile-time per step)
            int kk  = kb >> 3;            // 0,0,1,1,2,2
            v2f a, b;
            a.x = aBase[kb];
            a.y = aBase[kb + 1];
            b.x = bBase[icc * HPITCH + kk];
            b.y = bBase[(icc + 1) * HPITCH + kk];
            acc = __builtin_amdgcn_wmma_f32_16x16x4_f32(
                false, a, false, b, (short)0, acc, false, false);
        }

        float* hv = h + (size_t)view * OC1 * HPITCH
                      + (size_t)((ocT << 4) + (hi << 3)) * HPITCH
                      + 1 + p0 + row;
#pragma unroll
        for (int j = 0; j < 8; ++j) {
            float x = acc[j];
            hv[j * HPITCH] = 0.5f * x * (1.0f + erff(x * 0.70710678118654752f));
        }
    }
}

// =======================================================================
// Kernel 2: conv2 (112 -> 56 ch, k=3, pad=1) + bias, scatter into padded
// AoS xbuf[ray][c*8 + f].  WMMA f32 16x16x4, K = 336 in 84 steps.
// Padded h makes halo loads branch-free.  A-pointer clamped to row 55
// (garbage A rows only affect D rows we never store -> no EXEC branches).
// B addresses walk incrementally: r += 4  =>  addr += (k==2 ? 742 : 373).
// =======================================================================
__global__ __launch_bounds__(256) void k_conv2_scatter(
    const float* __restrict__ h,    // [V][112][372]
    const float* __restrict__ w2,   // [56][112][3]
    const float* __restrict__ b2,   // [56]
    float* __restrict__ xbuf)       // [NRAY][64]
{
    const int TILES = 4 * 23;                       // ceil(56/16) ocTiles * posTiles
    int wave = (blockIdx.x * 256 + threadIdx.x) >> 5;
    if (wave >= NVIEWS * TILES) return;
    int lane = threadIdx.x & 31;
    int view = wave / TILES;
    int t    = wave - view * TILES;
    int ocT  = t / 23;
    int pT   = t - ocT * 23;
    int p0   = pT << 4;
    int row  = lane & 15;
    int hi   = lane >> 4;
    int rA   = hi << 1;
    int oc   = (ocT << 4) + row;
    int ocl  = (oc < OC2) ? oc : (OC2 - 1);         // clamp, no branch needed

    v8f acc;
#pragma unroll
    for (int j = 0; j < 8; ++j) {
        int ocj = (ocT << 4) + (hi << 3) + j;
        acc[j] = b2[(ocj < OC2) ? ocj : (OC2 - 1)];
    }

    const float* hv = h + (size_t)view * OC1 * HPITCH;
    const float* aP = w2 + ocl * 336 + rA;          // 8B aligned (even offsets)

    // b.x starts at r0 = rA   : rA=0 -> (ic=0,k=0) ; rA=2 -> (ic=0,k=2)
    // b.y starts at r1 = rA+1 : 1    -> (ic=0,k=1) ; 3    -> (ic=1,k=0)
    int kx = (rA == 2) ? 2 : 0;
    int ky = (rA == 2) ? 0 : 1;
    int bx = p0 + row + kx;                         // ic=0 for both variants
    int by = ((rA == 2) ? HPITCH : 0) + p0 + row + ky;

#pragma unroll 2
    for (int step = 0; step < 84; ++step) {
        v2f a = *(const v2f*)aP;                    // global_load_b64, linear
        v2f b;
        b.x = hv[bx];
        b.y = hv[by];
        acc = __builtin_amdgcn_wmma_f32_16x16x4_f32(
            false, a, false, b, (short)0, acc, false, false);
        aP += 4;
        bx += (kx == 2) ? (2 * HPITCH - 2) : (HPITCH + 1);
        kx  = (kx == 2) ? 0 : kx + 1;
        by += (ky == 2) ? (2 * HPITCH - 2) : (HPITCH + 1);
        ky  = (ky == 2) ? 0 : ky + 1;
    }

    int u = p0 + row;
    size_t ray = (size_t)view * NDET + u;
#pragma unroll
    for (int j = 0; j < 8; ++j) {
        int ocj = (ocT << 4) + (hi << 3) + j;
        if (ocj < OC2) {
            int c = ocj / 7, f = ocj - c * 7;
            xbuf[ray * XRECORD + (c << 3) + f] = acc[j];
        }
    }
}

// =======================================================================
// Kernel 3 (dominant, memory-bound): per-m linear-interp gather + trig basis.
// One sincosf per m; cos2/sin2/cos3/sin3 by double/sum-angle recurrence;
// high side (w-1) by rotation with constants cos(1), sin(1).
// xbuf records are 256B-aligned -> 4x float4 per channel, all L2-resident.
// =======================================================================
__global__ __launch_bounds__(256) void k_gather_trig(
    const float* __restrict__ xbuf, // [NRAY][64]
    const float* __restrict__ idx,  // [MTOT]
    float* __restrict__ out)        // [CH][MTOT]
{
    int m = blockIdx.x * 256 + threadIdx.x;
    if (m >= MTOT) return;

    float t  = idx[m];
    float fl = floorf(t);
    int   i0 = (int)fl;
    float w  = t - fl;
    int   i1 = (int)ceilf(t);
    i1 = (i1 < NRAY - 1) ? i1 : (NRAY - 1);

    float s, c;
    sincosf(w, &s, &c);
    float c2 = __fmaf_rn(2.f * c, c, -1.f);
    float s2 = 2.f * s * c;
    float c3 = c * c2 - s * s2;
    float s3 = s * c2 + c * s2;

    const float C1 = 0.54030230586813972f;          // cos(1)
    const float S1 = 0.84147098480789651f;          // sin(1)
    float ch = c * C1 + s * S1;                     // cos(w-1)
    float sh = s * C1 - c * S1;                     // sin(w-1)
    float ch2 = __fmaf_rn(2.f * ch, ch, -1.f);
    float sh2 = 2.f * sh * ch;
    float ch3 = ch * ch2 - sh * sh2;
    float sh3 = sh * ch2 + ch * sh2;

    float wl = 1.f - w, wh = w;
    float L0 = wl,       L1 = c   * wl, L2 = s   * wl, L3 = c2  * wl,
          L4 = s2 * wl,  L5 = c3  * wl, L6 = s3  * wl;
    float H0 = wh,       H1 = ch  * wh, H2 = sh  * wh, H3 = ch2 * wh,
          H4 = sh2 * wh, H5 = ch3 * wh, H6 = sh3 * wh;

    const float4* lo = (const float4*)(xbuf + (size_t)i0 * XRECORD);
    const float4* hp = (const float4*)(xbuf + (size_t)i1 * XRECORD);
#pragma unroll
    for (int ch8 = 0; ch8 < 8; ++ch8) {
        float4 a = lo[ch8 * 2], b = lo[ch8 * 2 + 1];
        float4 d = hp[ch8 * 2], e = hp[ch8 * 2 + 1];
        float r = a.x * L0 + a.y * L1 + a.z * L2 + a.w * L3
                + b.x * L4 + b.y * L5 + b.z * L6
                + d.x * H0 + d.y * H1 + d.z * H2 + d.w * H3
                + e.x * H4 + e.y * H5 + e.z * H6;
        out[(size_t)ch8 * MTOT + m] = r;
    }
}

// =======================================================================
extern "C" void kernel_launch(void* const* d_in, const int* in_sizes, int n_in,
                              void* d_out, int out_size, void* d_ws, size_t ws_size,
                              hipStream_t stream)
{
    const float* in  = (const float*)d_in[0];   // [1,8,128,368]
    const float* idx = (const float*)d_in[1];   // [MTOT]
    const float* w1  = (const float*)d_in[2];   // [112,8,3]
    const float* b1  = (const float*)d_in[3];   // [112]
    const float* w2  = (const float*)d_in[4];   // [56,112,3]
    const float* b2  = (const float*)d_in[5];   // [56]
    float* out = (float*)d_out;

    float* h    = (float*)d_ws;                 // [128][112][372]
    float* xbuf = h + HFLOATS;                  // [47104][64]

    // zero pads/halos (graph-capture safe)
    hipMemsetAsync(d_ws, 0, (HFLOATS + XFLOATS) * sizeof(float), stream);

    // conv1+GELU: one block per view, 8 waves sweep 161 tiles
    k_conv1_gelu<<<dim3(NVIEWS), dim3(256), 0, stream>>>(in, w1, b1, h);
    // conv2+scatter: 128 * 92 = 11776 waves -> 1472 blocks
    k_conv2_scatter<<<dim3((NVIEWS * 4 * 23) / 8), dim3(256), 0, stream>>>(h, w2, b2, xbuf);
    // gather: one thread per m
    k_gather_trig<<<dim3(MTOT / 256), dim3(256), 0, stream>>>(xbuf, idx, out);
}